// ETGraphMaskModulator_25237227831476
// MI455X (gfx1250) — compile-verified
//
#include <hip/hip_runtime.h>

// Problem constants from the reference
#define B_ 8
#define N_ 256
#define D_ 64
#define H_ 8
#define P_ 4

typedef __attribute__((ext_vector_type(2))) float v2f;
typedef __attribute__((ext_vector_type(8))) float v8f;

// LDS layout (floats): weights first so their DS offsets are small immediates.
#define WOFF 0                   // sW[(di*3+dj)*64 + d], 576 floats
#define XOFF 576                 // sX[258][64]: x rows -1..256 zero-padded (local row r = global r-1)
#define SMEM_FLOATS (576 + 258 * D_)   // 17088 floats = 66.75 KB

// Fused kernel:
//   out[b,h,i,j] = (conv_b[h] + sum_{di,d} xpad[b,i+di-1,d] * Z_di[j,d]) * edge_heads
//   Z_di[j,d]    = sum_{dj} conv_w[h,d,di,dj] * xpad[b,j+dj-1,d]
//   edge_heads   = sum_p edge_features[b,i,j,p] * edge_w[h,p]
// Each wave owns ONE 16-col j-tile for the whole block: it builds its per-lane
// B operands Z_di (48 x v2f, 96 VGPRs) ONCE, then sweeps 8 i-tiles where each
// K-step is only 3 ds loads + 3 V_WMMA_F32_16X16X4_F32 (exact fp32 numerics).
// Two accumulators alternate per WMMA to halve the XDL RAW chain depth.
__global__ __launch_bounds__(256) void etgmm_fused_kernel(
    const float* __restrict__ x,       // [B,N,D]
    const float* __restrict__ edge,    // [B,N,N,P]
    const float* __restrict__ conv_w,  // [H,D,3,3]
    const float* __restrict__ conv_b,  // [H]
    const float* __restrict__ edge_w,  // [H,P]
    float* __restrict__ out)           // [B,H,N,N]
{
    __shared__ float smem[SMEM_FLOATS];

    const int tid  = threadIdx.x;
    const int lane = tid & 31;
    const int wave = tid >> 5;
    const int n    = lane & 15;   // A row m / B col n / C col within tile
    const int hi   = lane >> 4;
    const int kb   = hi << 1;     // per-lane K base (0 or 2) for f32 WMMA A/B layout

    const int bh    = blockIdx.z;  // bh = b*H + h
    const int h     = bh & 7;
    const int b     = bh >> 3;
    const int jbase = blockIdx.x << 7;        // 128-col half per block
    const int j0    = jbase + (wave << 4);    // this wave's 16-col tile (for all i)
    const int ibase = blockIdx.y << 7;        // 128-row half swept by this block

    const float* __restrict__ xb = x + b * (N_ * D_);
    const v2f zero2 = {0.f, 0.f};

    // --- Stage conv_w[h,:,:,:] at LDS offset 0 (small, foldable DS offsets) ---
    for (int t = tid; t < 9 * D_; t += 256) {
        const int d = t / 9;
        const int r = t - d * 9;                 // r = di*3 + dj
        smem[WOFF + r * D_ + d] = conv_w[(h * D_ + d) * 9 + r];
    }
    // --- Stage full zero-padded x image: 258 rows, b64 chunks ---
    for (int t = tid; t < 258 * (D_ / 2); t += 256) {
        const int row = t >> 5;
        const int dd  = (t & 31) << 1;
        const int g   = row - 1;
        const v2f v = (g >= 0 && g < N_) ? *(const v2f*)(xb + g * D_ + dd) : zero2;
        *(v2f*)&smem[XOFF + row * D_ + dd] = v;
    }
    __syncthreads();

    // --- Build this wave's B operands once: zreg[t*3+di] = Z_di[j0+n, 4t+kb .. +1] ---
    // Z_di[j,d] = sum_dj w[h,d,di,dj] * xpad[j+dj-1,d]; global row j0+n+dj-1 -> local j0+n+dj.
    v2f zreg[48];
    {
        const float* __restrict__ pJ = &smem[XOFF + (j0 + n) * D_];
        #pragma unroll
        for (int t = 0; t < 16; ++t) {
            const int d0 = (t << 2) + kb;
            const v2f xj0 = *(const v2f*)(pJ + 0 * D_ + d0);
            const v2f xj1 = *(const v2f*)(pJ + 1 * D_ + d0);
            const v2f xj2 = *(const v2f*)(pJ + 2 * D_ + d0);
            #pragma unroll
            for (int di = 0; di < 3; ++di) {
                const v2f w0 = *(const v2f*)&smem[WOFF + (di * 3 + 0) * D_ + d0];
                const v2f w1 = *(const v2f*)&smem[WOFF + (di * 3 + 1) * D_ + d0];
                const v2f w2 = *(const v2f*)&smem[WOFF + (di * 3 + 2) * D_ + d0];
                v2f z;
                z.x = w0.x * xj0.x + w1.x * xj1.x + w2.x * xj2.x;
                z.y = w0.y * xj0.y + w1.y * xj1.y + w2.y * xj2.y;
                zreg[t * 3 + di] = z;
            }
        }
    }

    const float  bias = conv_b[h];
    const float4 ew   = *(const float4*)(edge_w + h * P_);

    // --- Sweep 8 i-tiles, reusing the register-resident B operands ---
    #pragma unroll 1
    for (int it = 0; it < 8; ++it) {
        const int i0 = ibase + (it << 4);

        // Prefetch this tile's edge_features: one base pointer, row stride folds
        // into the 24-bit instruction offset (rows are N_*P_*4 = 4096 B apart).
        {
            const char* pe =
                (const char*)(edge + ((b * N_ + i0 + (hi << 3)) * N_ + j0 + n) * P_);
            #pragma unroll
            for (int r = 0; r < 8; ++r)
                __builtin_prefetch(pe + r * (N_ * P_ * 4), 0, 1);
        }

        v8f acc0 = {};  // even-index WMMAs
        v8f acc1 = {};  // odd-index WMMAs (chain depth halved)
        // A row for (n,di): global i0+n+di-1 -> local i0+n+di.
        const float* __restrict__ pA = &smem[XOFF + (i0 + n) * D_];

        #pragma unroll
        for (int t = 0; t < 16; ++t) {
            const int d0 = (t << 2) + kb;
            const v2f xi0 = *(const v2f*)(pA + 0 * D_ + d0);
            const v2f xi1 = *(const v2f*)(pA + 1 * D_ + d0);
            const v2f xi2 = *(const v2f*)(pA + 2 * D_ + d0);
            // D = A(16x4,f32) x B(4x16,f32) + C ; alternate accumulators per WMMA
            if ((t & 1) == 0) {
                acc0 = __builtin_amdgcn_wmma_f32_16x16x4_f32(
                    false, xi0, false, zreg[t * 3 + 0], (short)0, acc0, false, false);
                acc1 = __builtin_amdgcn_wmma_f32_16x16x4_f32(
                    false, xi1, false, zreg[t * 3 + 1], (short)0, acc1, false, false);
                acc0 = __builtin_amdgcn_wmma_f32_16x16x4_f32(
                    false, xi2, false, zreg[t * 3 + 2], (short)0, acc0, false, false);
            } else {
                acc1 = __builtin_amdgcn_wmma_f32_16x16x4_f32(
                    false, xi0, false, zreg[t * 3 + 0], (short)0, acc1, false, false);
                acc0 = __builtin_amdgcn_wmma_f32_16x16x4_f32(
                    false, xi1, false, zreg[t * 3 + 1], (short)0, acc0, false, false);
                acc1 = __builtin_amdgcn_wmma_f32_16x16x4_f32(
                    false, xi2, false, zreg[t * 3 + 2], (short)0, acc1, false, false);
            }
        }

        // Epilogue: bias, edge projection (P=4 contiguous -> float4), modulate, store.
        #pragma unroll
        for (int r = 0; r < 8; ++r) {
            const int i = i0 + r + (hi << 3);  // C layout: VGPR r = rows r (lanes 0-15), r+8 (lanes 16-31)
            const int j = j0 + n;
            const float4 ef = *(const float4*)(edge + ((b * N_ + i) * N_ + j) * P_);
            const float eh  = ef.x * ew.x + ef.y * ew.y + ef.z * ew.z + ef.w * ew.w;
            out[(bh * N_ + i) * N_ + j] = (acc0[r] + acc1[r] + bias) * eh;
        }
    }
}

extern "C" void kernel_launch(void* const* d_in, const int* in_sizes, int n_in,
                              void* d_out, int out_size, void* d_ws, size_t ws_size,
                              hipStream_t stream) {
    const float* x      = (const float*)d_in[0];  // [B,N,D]
    const float* edge   = (const float*)d_in[1];  // [B,N,N,P]
    const float* conv_w = (const float*)d_in[2];  // [H,D,3,3]
    const float* conv_b = (const float*)d_in[3];  // [H]
    const float* edge_w = (const float*)d_in[4];  // [H,P]
    float* out = (float*)d_out;                   // [B,H,N,N]

    // grid: x = 2 j-halves, y = 2 i-halves, z = B*H  (256 blocks, 8 waves each).
    // Each block sweeps 8 i-tiles internally, reusing register-resident Z.
    dim3 grid(2, 2, B_ * H_);
    dim3 block(256);
    etgmm_fused_kernel<<<grid, block, 0, stream>>>(x, edge, conv_w, conv_b, edge_w, out);
}